// Communication_88441966559569
// MI455X (gfx1250) — compile-verified
//
#include <hip/hip_runtime.h>
#include <cstdint>

#define BATCH 4
#define HGT 192
#define WID 192
#define HW (HGT * WID)                 // 36864 pixels per batch
#define NPIX (BATCH * HW)              // 147456
#define CCH 90                         // 10 vox + 64 feat + 16 det
#define SPARSE_ELEMS ((size_t)BATCH * CCH * HW)  // 13,271,040 floats

// ---------------------------------------------------------------------------
// Kernel 1: per-pixel argmax over the 3 utility channels, build 32-bit
// order-preserving sort key + meta byte (class in bits[1:0], valid in bit 2).
// ---------------------------------------------------------------------------
__global__ __launch_bounds__(256) void prep_kernel(const float* __restrict__ util,
                                                   uint32_t* __restrict__ fkey,
                                                   uint8_t* __restrict__ meta)
{
    int i = blockIdx.x * 256 + threadIdx.x;
    if (i >= NPIX) return;
    float u0 = util[3 * i + 0];
    float u1 = util[3 * i + 1];
    float u2 = util[3 * i + 2];
    float bu = u0; int g = 0;
    if (u1 > bu) { bu = u1; g = 1; }     // strict > => first-max wins (jnp.argmax)
    if (u2 > bu) { bu = u2; g = 2; }
    uint32_t bits = __float_as_uint(bu);
    uint32_t key  = (bits & 0x80000000u) ? ~bits : (bits | 0x80000000u); // monotone map
    fkey[i] = key;
    meta[i] = (uint8_t)((unsigned)g | ((bu > 0.0f) ? 4u : 0u));
}

// ---------------------------------------------------------------------------
// Kernel 2: exact greedy selection WITHOUT sorting.
// The greedy scan over descending utility has monotone "remaining" budget, so
// each cost class is selected on a prefix of the sorted order. We find the
// three prefix-boundary keys (for caps budget-10, budget-5, budget-2, with
// nested cohorts and carried base usage) via weighted 256-ary radix-select
// over 48-bit keys K = (fkey << 16) | (0xFFFF - idx). One workgroup per batch.
// ---------------------------------------------------------------------------
__global__ __launch_bounds__(1024) void select_kernel(const uint32_t* __restrict__ fkey,
                                                      const uint8_t* __restrict__ meta,
                                                      const int* __restrict__ budget_i,
                                                      float* __restrict__ sel_out)
{
    __shared__ unsigned bins[256];
    __shared__ int      shChosen;
    __shared__ unsigned shAcc;
    __shared__ unsigned shWx;

    const int b = blockIdx.x;            // batch
    const int off = b * HW;
    const unsigned tid = threadIdx.x;

    const float budget = (float)((double)budget_i[0] / (double)BATCH);  // 50000
    const float caps[3] = { budget - 10.0f, budget - 5.0f, budget - 2.0f };

    unsigned long long bound = ~0ull;    // cohort upper bound (exclusive)
    unsigned baseUsage = 0;              // usage accumulated before cohort
    unsigned long long Xarr[3];          // selection thresholds per phase

    for (int ph = 0; ph < 3; ++ph) {
        const float cap = caps[ph];
        unsigned long long X;
        if ((float)baseUsage > cap) {
            // Budget already exceeded before this cohort begins.
            X = bound;
        } else {
            unsigned long long prefix = 0;
            unsigned curBase = baseUsage;
            bool found = true;
            for (int lvl = 0; lvl < 6; ++lvl) {
                const int shift = 40 - 8 * lvl;
                __syncthreads();
                for (unsigned i = tid; i < 256u; i += 1024u) bins[i] = 0;
                __syncthreads();
                for (int t = (int)tid; t < HW; t += 1024) {
                    const unsigned m = meta[off + t];
                    unsigned w = 0;
                    if (m & 4u) {                       // valid (u > 0)
                        const unsigned g = m & 3u;
                        if      (ph == 0) w = (g == 0u) ? 10u : (g == 1u) ? 5u : 2u;
                        else if (ph == 1) w = (g == 1u) ? 5u  : (g == 2u) ? 2u : 0u;
                        else              w = (g == 2u) ? 2u  : 0u;
                    }
                    if (!w) continue;
                    const unsigned long long K =
                        ((unsigned long long)fkey[off + t] << 16) |
                        (unsigned)(0xFFFF - t);
                    if (K >= bound) continue;           // outside cohort
                    if ((K >> (shift + 8)) != (prefix >> (shift + 8))) continue;
                    atomicAdd(&bins[(unsigned)(K >> shift) & 0xFFu], w);
                }
                __syncthreads();
                if (tid == 0) {
                    unsigned acc = curBase;
                    int chosen = -1; unsigned wx = 0;
                    for (int bb = 255; bb >= 0; --bb) { // descending keys first
                        const unsigned wb = bins[bb];
                        if (wb != 0u && (float)(acc + wb) > cap) { chosen = bb; wx = wb; break; }
                        acc += wb;
                    }
                    shChosen = chosen; shAcc = acc; shWx = wx;
                }
                __syncthreads();
                const int chosen = shChosen;
                curBase = shAcc;
                if (chosen < 0) { found = false; break; }  // no crossing (lvl 0 only)
                prefix |= (unsigned long long)chosen << shift;
                if (lvl == 5) curBase += shWx;             // include crossing item
            }
            if (found) { X = prefix; baseUsage = curBase; }
            else       { X = 0ull;   baseUsage = curBase; }  // everything fits
        }
        Xarr[ph] = X;
        bound = X;
    }

    __syncthreads();
    // Item selected <=> valid && K >= threshold of its class's phase.
    for (int t = (int)tid; t < HW; t += 1024) {
        const unsigned m = meta[off + t];
        const unsigned g = m & 3u;
        const unsigned long long K =
            ((unsigned long long)fkey[off + t] << 16) | (unsigned)(0xFFFF - t);
        const bool sel = (m & 4u) && (K >= Xarr[g]);
        sel_out[off + t] = sel ? (float)g : -1.0f;
    }
}

// ---------------------------------------------------------------------------
// Kernel 3: bandwidth-bound masked copy (the ~106 MB stream).
// Each block stages a 1024-pixel selection tile into LDS via the CDNA5 async
// global->LDS path (ASYNCcnt), then streams 30 channels with B128 accesses.
// Grid = BATCH(4) * tiles(36) * channel-groups(3) = 432 blocks of 256 threads.
// ---------------------------------------------------------------------------
__global__ __launch_bounds__(256) void scatter_kernel(const float* __restrict__ collab,
                                                      const float* __restrict__ sel,
                                                      float* __restrict__ out)
{
    __shared__ float s_sel[1024];
    const int blk  = blockIdx.x;
    const int b    = blk / 108;
    const int rem  = blk % 108;
    const int tile = rem / 3;
    const int cg   = rem % 3;
    const int tid  = threadIdx.x;
    const int pix0 = tile * 1024;

    // --- async stage of the selection tile into LDS (one B128 per lane) ---
    {
        const float* sp = sel + (size_t)b * HW + pix0 + tid * 4;
        typedef __attribute__((address_space(3))) float lds_float;
        lds_float* lp = (lds_float*)(&s_sel[tid * 4]);
        unsigned lds_off = (unsigned)(unsigned long long)lp;   // LDS byte address
        unsigned long long gaddr = (unsigned long long)(uintptr_t)sp;
        asm volatile("global_load_async_to_lds_b128 %0, %1, off"
                     :: "v"(lds_off), "v"(gaddr)
                     : "memory");
#if __has_builtin(__builtin_amdgcn_s_wait_asynccnt)
        __builtin_amdgcn_s_wait_asynccnt(0);
#else
        asm volatile("s_wait_asynccnt 0" ::: "memory");
#endif
    }
    __syncthreads();

    const float s0 = s_sel[tid * 4 + 0];
    const float s1 = s_sel[tid * 4 + 1];
    const float s2 = s_sel[tid * 4 + 2];
    const float s3 = s_sel[tid * 4 + 3];

    const int c0 = cg * 30;
    const size_t basei = (((size_t)b * CCH + c0) * HW) + pix0 + tid * 4;
    const float* src = collab + basei;
    float* dst = out + basei;

#pragma unroll 5
    for (int c = 0; c < 30; ++c) {
        const int ch = c0 + c;
        const float cls = (ch < 10) ? 0.0f : (ch < 74) ? 1.0f : 2.0f;
        const float4 v = *(const float4*)(src + (size_t)c * HW);
        float4 r;
        r.x = (s0 == cls) ? v.x : 0.0f;
        r.y = (s1 == cls) ? v.y : 0.0f;
        r.z = (s2 == cls) ? v.z : 0.0f;
        r.w = (s3 == cls) ? v.w : 0.0f;
        *(float4*)(dst + (size_t)c * HW) = r;
    }
}

// ---------------------------------------------------------------------------
extern "C" void kernel_launch(void* const* d_in, const int* in_sizes, int n_in,
                              void* d_out, int out_size, void* d_ws, size_t ws_size,
                              hipStream_t stream)
{
    (void)in_sizes; (void)n_in; (void)out_size; (void)ws_size;
    const float* collab = (const float*)d_in[0];  // (4, 90, 192, 192) f32
    const float* util   = (const float*)d_in[1];  // (4, 192, 192, 3) f32
    const int*   budget = (const int*)d_in[2];    // scalar int

    float* out_sparse = (float*)d_out;                 // (4, 90, 192, 192)
    float* out_sel    = (float*)d_out + SPARSE_ELEMS;  // (4, 192, 192)

    uint32_t* fkey = (uint32_t*)d_ws;                        // NPIX u32
    uint8_t*  meta = (uint8_t*)d_ws + (size_t)NPIX * 4;      // NPIX u8

    prep_kernel<<<(NPIX + 255) / 256, 256, 0, stream>>>(util, fkey, meta);
    select_kernel<<<BATCH, 1024, 0, stream>>>(fkey, meta, budget, out_sel);
    scatter_kernel<<<BATCH * 36 * 3, 256, 0, stream>>>(collab, out_sel, out_sparse);
}